// Layer_34428457845564
// MI455X (gfx1250) — compile-verified
//
#include <hip/hip_runtime.h>
#include <math.h>

#define NN 100000
#define EE 1600000
#define C_IN 32
#define C_OUT 64
#define KB 8           // basis functions
#define LD1 640        // 8*64 + 64 (root1) + 64 (skip)
#define LD2 576        // 8*64 + 64 (root2)

typedef float v2f __attribute__((ext_vector_type(2)));
typedef float v4f __attribute__((ext_vector_type(4)));
typedef float v8f __attribute__((ext_vector_type(8)));

// ---------------- packing kernels ----------------
__global__ void pack_w1(const float* __restrict__ W1, const float* __restrict__ root1,
                        const float* __restrict__ Wskip, float* __restrict__ wp) {
    int i = blockIdx.x * blockDim.x + threadIdx.x;
    if (i >= C_IN * LD1) return;
    int r = i / LD1, c = i % LD1;
    float v;
    if (c < 512)       v = W1[(c >> 6) * (C_IN * 64) + r * 64 + (c & 63)];
    else if (c < 576)  v = root1[r * 64 + (c - 512)];
    else               v = Wskip[r * 64 + (c - 576)];
    wp[i] = v;
}

__global__ void pack_w2(const float* __restrict__ W2, const float* __restrict__ root2,
                        float* __restrict__ wp) {
    int i = blockIdx.x * blockDim.x + threadIdx.x;
    if (i >= C_OUT * LD2) return;
    int r = i / LD2, c = i % LD2;
    float v;
    if (c < 512) v = W2[(c >> 6) * (C_OUT * 64) + r * 64 + (c & 63)];
    else         v = root2[r * 64 + (c - 512)];
    wp[i] = v;
}

// ---------------- WMMA f32 GEMM, async-LDS staged B ----------------
// C(M x ldb) = A(M x K) @ B(K x ldb). 320-thread block = 10 waves.
// Block covers 160 rows x 64 cols; B panel (K x 64) staged once into LDS via
// gfx1250 GLOBAL_LOAD_ASYNC_TO_LDS_B32 (ASYNCcnt), then all waves feed WMMA
// from ds_load. M = 100000 = 625 * 160, so no tails; EXEC all-ones at WMMAs.
// Fragment layouts (ISA 7.12.2, wave32):
//   A 16x4: lane = m + 16*(k>=2); vgpr j holds K = 2*half + j -> contiguous float2
//   B 4x16: vgpr j holds row K = 2*half + j, N = lane&15
//   C/D:    vgpr v holds M = v + 8*half, N = lane&15
template <int K>
__global__ __launch_bounds__(320) void gemm_wmma_f32(const float* __restrict__ A,
                                                     const float* __restrict__ Bg,
                                                     float* __restrict__ C, int ldb) {
    __shared__ float bp[K * 64];
    const int tid  = threadIdx.x;
    const int col0 = blockIdx.y * 64;

    // stage B panel (K x 64) into LDS with async copies
    for (int i = tid; i < K * 64; i += 320) {
        const float* gp = Bg + (size_t)(i >> 6) * ldb + col0 + (i & 63);
        unsigned lds = (unsigned)(uintptr_t)&bp[i];
        unsigned long long ga = (unsigned long long)(uintptr_t)gp;
        asm volatile("global_load_async_to_lds_b32 %0, %1, off"
                     :: "v"(lds), "v"(ga) : "memory");
    }
    asm volatile("s_wait_asynccnt 0x0" ::: "memory");
    __syncthreads();

    const int wave = tid >> 5;
    const int lane = tid & 31;
    const int half = lane >> 4;
    const int l16  = lane & 15;
    const long row = (long)blockIdx.x * 160 + wave * 16 + l16;

    v8f acc0 = {}, acc1 = {}, acc2 = {}, acc3 = {};
    const float* Arow = A + row * K;

#pragma unroll
    for (int kk = 0; kk < K; kk += 4) {
        v2f a = *(const v2f*)(Arow + kk + 2 * half);
        const float* bk = &bp[(kk + 2 * half) * 64 + l16];   // row stride 64 in panel
        v2f b0 = { bk[0],  bk[64 + 0]  };
        v2f b1 = { bk[16], bk[64 + 16] };
        v2f b2 = { bk[32], bk[64 + 32] };
        v2f b3 = { bk[48], bk[64 + 48] };
        acc0 = __builtin_amdgcn_wmma_f32_16x16x4_f32(false, a, false, b0, (short)0, acc0, false, false);
        acc1 = __builtin_amdgcn_wmma_f32_16x16x4_f32(false, a, false, b1, (short)0, acc1, false, false);
        acc2 = __builtin_amdgcn_wmma_f32_16x16x4_f32(false, a, false, b2, (short)0, acc2, false, false);
        acc3 = __builtin_amdgcn_wmma_f32_16x16x4_f32(false, a, false, b3, (short)0, acc3, false, false);
    }

    float* Crow = C + ((long)blockIdx.x * 160 + wave * 16 + 8 * half) * ldb + col0 + l16;
#pragma unroll
    for (int v = 0; v < 8; ++v) {
        float* p = Crow + (size_t)v * ldb;
        p[0] = acc0[v]; p[16] = acc1[v]; p[32] = acc2[v]; p[48] = acc3[v];
    }
}

// ---------------- edge message + scatter ----------------
// 16 threads per edge, 4 channels each (b128 gathers).
// msg[e,o] = sum_k basis(e,k) * Y[src, k*64 + o]; atomic f32 add into agg[dst].
__global__ void edge_msg(const float* __restrict__ Y, int ldy,
                         const int* __restrict__ ei, const float* __restrict__ attr,
                         float* __restrict__ agg, float* __restrict__ cnt, int addCnt) {
    int e = blockIdx.x * (blockDim.x >> 4) + (threadIdx.x >> 4);
    if (e >= EE) return;
    int o4 = (threadIdx.x & 15) * 4;
    int s = ei[e];
    int d = ei[EE + e];
    float u0 = attr[e * 3 + 0], u1 = attr[e * 3 + 1], u2 = attr[e * 3 + 2];
    float f0[2] = { 1.f - u0, u0 };
    float f1[2] = { 1.f - u1, u1 };
    float f2[2] = { 1.f - u2, u2 };
    const float* y = Y + (size_t)s * ldy + o4;
    v4f t = {};
#pragma unroll
    for (int k = 0; k < KB; ++k) {
        float bk = f0[k & 1] * f1[(k >> 1) & 1] * f2[(k >> 2) & 1];
        v4f yv = *(const v4f*)(y + k * 64);
        t += bk * yv;
    }
    float* a = agg + (size_t)d * 64 + o4;
    atomicAdd(a + 0, t.x);
    atomicAdd(a + 1, t.y);
    atomicAdd(a + 2, t.z);
    atomicAdd(a + 3, t.w);
    if (addCnt && (threadIdx.x & 15) == 0) atomicAdd(&cnt[d], 1.0f);
}

// ---------------- combine / BN / activation ----------------
__global__ void combine1(const float* __restrict__ agg, const float* __restrict__ cnt,
                         const float* __restrict__ Y, float* __restrict__ h,
                         float* __restrict__ skip) {
    int i = blockIdx.x * blockDim.x + threadIdx.x;
    if (i >= NN * 64) return;
    int n = i >> 6, c = i & 63;
    const float* yr = Y + (size_t)n * LD1;
    float inv = 1.0f / fmaxf(cnt[n], 1.0f);
    h[i]    = agg[i] * inv + yr[512 + c];
    skip[i] = yr[576 + c];
}

__global__ void combine2(const float* __restrict__ agg, const float* __restrict__ cnt,
                         const float* __restrict__ Y, float* __restrict__ h) {
    int i = blockIdx.x * blockDim.x + threadIdx.x;
    if (i >= NN * 64) return;
    int n = i >> 6, c = i & 63;
    float inv = 1.0f / fmaxf(cnt[n], 1.0f);
    h[i] = agg[i] * inv + Y[(size_t)n * LD2 + 512 + c];
}

// per-channel sums in f64; stride is a multiple of 64 so each thread's channel is fixed
__global__ void bn_stats(const float* __restrict__ X, double* __restrict__ dsum,
                         double* __restrict__ dsq) {
    int stride = gridDim.x * blockDim.x;
    int i = blockIdx.x * blockDim.x + threadIdx.x;
    int c = i & 63;
    double s = 0.0, q = 0.0;
    for (; i < NN * 64; i += stride) {
        float v = X[i];
        s += v;
        q += (double)v * (double)v;
    }
    atomicAdd(&dsum[c], s);
    atomicAdd(&dsq[c], q);
}

__global__ void bn_finalize(const double* __restrict__ dsum, const double* __restrict__ dsq,
                            float* __restrict__ musr) {
    int c = threadIdx.x;  // 64 threads
    const double invN = 1.0 / (double)NN;
    double mu  = dsum[c] * invN;
    double var = dsq[c] * invN - mu * mu;
    musr[c]      = (float)mu;
    musr[64 + c] = rsqrtf((float)var + 1e-5f);
}

__global__ void apply_bn(float* __restrict__ X, const float* __restrict__ musr,
                         const float* __restrict__ g, const float* __restrict__ b, int doElu) {
    int i = blockIdx.x * blockDim.x + threadIdx.x;
    if (i >= NN * 64) return;
    int c = i & 63;
    float v = (X[i] - musr[c]) * musr[64 + c] * g[c] + b[c];
    if (doElu) v = v > 0.f ? v : expf(v) - 1.f;
    X[i] = v;
}

__global__ void final_out(const float* __restrict__ h, const float* __restrict__ skip,
                          const float* __restrict__ musr, const float* __restrict__ g,
                          const float* __restrict__ b, float* __restrict__ out) {
    int i = blockIdx.x * blockDim.x + threadIdx.x;
    if (i >= NN * 64) return;
    int c = i & 63;
    float v = (h[i] - musr[c]) * musr[64 + c] * g[c] + b[c] + skip[i];
    out[i] = v > 0.f ? v : expf(v) - 1.f;
}

// ---------------- launcher ----------------
extern "C" void kernel_launch(void* const* d_in, const int* in_sizes, int n_in,
                              void* d_out, int out_size, void* d_ws, size_t ws_size,
                              hipStream_t stream) {
    const float* x     = (const float*)d_in[0];
    const int*   ei    = (const int*)  d_in[1];
    const float* attr  = (const float*)d_in[2];
    const float* W1    = (const float*)d_in[3];
    const float* root1 = (const float*)d_in[4];
    const float* g1    = (const float*)d_in[5];
    const float* b1    = (const float*)d_in[6];
    const float* W2    = (const float*)d_in[7];
    const float* root2 = (const float*)d_in[8];
    const float* g2    = (const float*)d_in[9];
    const float* b2    = (const float*)d_in[10];
    const float* Wsk   = (const float*)d_in[11];
    const float* gsk   = (const float*)d_in[12];
    const float* bsk   = (const float*)d_in[13];
    float* out = (float*)d_out;

    // workspace layout
    double* dsum = (double*)d_ws;           // 64
    double* dsq  = dsum + 64;               // 64
    float* musr0 = (float*)(dsum + 128);    // 128 (h BN)
    float* musr1 = musr0 + 128;             // 128 (skip BN)
    float* musr2 = musr1 + 128;             // 128 (h2 BN)
    float* wp1   = musr2 + 128;             // 32*640
    float* wp2   = wp1 + C_IN * LD1;        // 64*576
    float* Y     = wp2 + C_OUT * LD2;       // NN*640 (reused as NN*576)
    float* agg   = Y + (size_t)NN * LD1;    // NN*64
    float* cnt   = agg + (size_t)NN * 64;   // NN
    float* hbuf  = cnt + NN;                // NN*64
    float* skip  = hbuf + (size_t)NN * 64;  // NN*64

    const int TPB = 256;
    const int nEl = NN * 64;
    const int elBlocks = (nEl + TPB - 1) / TPB;
    const int edgeBlocks = EE / 16;         // 16 edges per 256-thread block

    // ---- layer 1 ----
    pack_w1<<<(C_IN * LD1 + TPB - 1) / TPB, TPB, 0, stream>>>(W1, root1, Wsk, wp1);
    {
        dim3 g(NN / 160, LD1 / 64);
        gemm_wmma_f32<C_IN><<<g, 320, 0, stream>>>(x, wp1, Y, LD1);
    }
    hipMemsetAsync(agg, 0, (size_t)nEl * sizeof(float), stream);
    hipMemsetAsync(cnt, 0, (size_t)NN * sizeof(float), stream);
    edge_msg<<<edgeBlocks, TPB, 0, stream>>>(Y, LD1, ei, attr, agg, cnt, 1);
    combine1<<<elBlocks, TPB, 0, stream>>>(agg, cnt, Y, hbuf, skip);

    hipMemsetAsync(dsum, 0, 128 * sizeof(double), stream);
    bn_stats<<<256, TPB, 0, stream>>>(hbuf, dsum, dsq);
    bn_finalize<<<1, 64, 0, stream>>>(dsum, dsq, musr0);
    hipMemsetAsync(dsum, 0, 128 * sizeof(double), stream);
    bn_stats<<<256, TPB, 0, stream>>>(skip, dsum, dsq);
    bn_finalize<<<1, 64, 0, stream>>>(dsum, dsq, musr1);

    apply_bn<<<elBlocks, TPB, 0, stream>>>(hbuf, musr0, g1, b1, 1);   // h = elu(bn(h_pre))
    apply_bn<<<elBlocks, TPB, 0, stream>>>(skip, musr1, gsk, bsk, 0); // skip = bn(x@Wsk)

    // ---- layer 2 ----
    pack_w2<<<(C_OUT * LD2 + TPB - 1) / TPB, TPB, 0, stream>>>(W2, root2, wp2);
    {
        dim3 g(NN / 160, LD2 / 64);
        gemm_wmma_f32<C_OUT><<<g, 320, 0, stream>>>(hbuf, wp2, Y, LD2);
    }
    hipMemsetAsync(agg, 0, (size_t)nEl * sizeof(float), stream);
    edge_msg<<<edgeBlocks, TPB, 0, stream>>>(Y, LD2, ei, attr, agg, cnt, 0);
    combine2<<<elBlocks, TPB, 0, stream>>>(agg, cnt, Y, hbuf);

    hipMemsetAsync(dsum, 0, 128 * sizeof(double), stream);
    bn_stats<<<256, TPB, 0, stream>>>(hbuf, dsum, dsq);
    bn_finalize<<<1, 64, 0, stream>>>(dsum, dsq, musr2);

    final_out<<<elBlocks, TPB, 0, stream>>>(hbuf, skip, musr2, g2, b2, out);
}